// GCNConv_30477087933050
// MI455X (gfx1250) — compile-verified
//
#include <hip/hip_runtime.h>
#include <math.h>

typedef __attribute__((ext_vector_type(2))) float v2f;
typedef __attribute__((ext_vector_type(8))) float v8f;

#define D_FEAT 128

// --- 1. deg[n] = 1.0 (self-loop weight) ---
__global__ void k_init_deg(float* __restrict__ deg, int n) {
    int i = blockIdx.x * blockDim.x + threadIdx.x;
    if (i < n) deg[i] = 1.0f;
}

// --- 2. deg[dst[e]] += ew[e] ---
__global__ void k_deg_accum(float* __restrict__ deg, const int* __restrict__ dst,
                            const float* __restrict__ ew, int E) {
    int e = blockIdx.x * blockDim.x + threadIdx.x;
    if (e < E) atomicAdd(&deg[dst[e]], ew[e]);
}

// --- 3. dinv[n] = deg>0 ? rsqrt(deg) : 0 (in place) ---
__global__ void k_dinv(float* __restrict__ deg, int n) {
    int i = blockIdx.x * blockDim.x + threadIdx.x;
    if (i < n) {
        float d = deg[i];
        deg[i] = (d > 0.0f) ? rsqrtf(d) : 0.0f;
    }
}

// --- 4. h = x @ W^T via V_WMMA_F32_16X16X4_F32 ---
// x: [N,128] row-major, W: [128,128] row-major ([n_out][k]), h: [N,128]
// One wave computes one 16x16 tile of h. 8 waves/block = all 8 N-tiles.
__global__ void k_gemm_wmma(const float* __restrict__ x, const float* __restrict__ W,
                            float* __restrict__ h, int N) {
    const int lane  = threadIdx.x & 31;
    const int wave  = threadIdx.x >> 5;      // 0..7 -> N tile
    const int Mbase = blockIdx.x * 16;
    const int Nbase = wave * 16;
    const int half  = lane >> 4;             // 0: lanes 0-15, 1: lanes 16-31
    const int l15   = lane & 15;

    // A fragment rows (clamped so EXEC stays all-1 even on a ragged last tile)
    int mrow = Mbase + l15; if (mrow >= N) mrow = N - 1;
    const float* __restrict__ arow = x + (size_t)mrow * D_FEAT;
    // B fragment: B[k][n] = W[n][k]; W row-major => same indexing as A
    const float* __restrict__ brow = W + (size_t)(Nbase + l15) * D_FEAT;

    v8f c = {};
#pragma unroll
    for (int k = 0; k < D_FEAT; k += 4) {
        const int kk = k + 2 * half;         // ISA A/B layout: K-halves split at lane 16
        v2f a = *reinterpret_cast<const v2f*>(arow + kk);
        v2f b = *reinterpret_cast<const v2f*>(brow + kk);
        // 8 args: (neg_a, A, neg_b, B, c_mod, C, reuse_a, reuse_b)
        c = __builtin_amdgcn_wmma_f32_16x16x4_f32(false, a, false, b, (short)0, c,
                                                  false, false);
    }

    // C/D layout: VGPR v -> row (v) for lanes 0-15, row (v+8) for lanes 16-31
    float* __restrict__ hout = h + (size_t)Mbase * D_FEAT + Nbase;
#pragma unroll
    for (int v = 0; v < 8; ++v) {
        int m = v + 8 * half;
        if (Mbase + m < N) hout[(size_t)m * D_FEAT + l15] = c[v];
    }
}

// --- 5. out[n][d] = b[d] + h[n][d] * dinv[n]^2  (bias + self-loop message) ---
__global__ void k_out_init(float* __restrict__ out, const float* __restrict__ h,
                           const float* __restrict__ dinv, const float* __restrict__ bias,
                           int total) {
    int i = blockIdx.x * blockDim.x + threadIdx.x;
    if (i < total) {
        int node = i >> 7;        // /128
        int d    = i & 127;
        float di = dinv[node];
        out[i] = bias[d] + h[i] * di * di;
    }
}

// --- 6. scatter: out[dst] += h[src] * (dinv[src]*ew*dinv[dst]) ---
// One wave per edge (grid-stride). Lane handles 4 contiguous feats via float4
// gather (coalesced 512B per wave, L2-resident) + 4 f32 atomics.
__global__ void k_scatter(float* __restrict__ out, const float* __restrict__ h,
                          const float* __restrict__ dinv, const int* __restrict__ src,
                          const int* __restrict__ dst, const float* __restrict__ ew,
                          int E) {
    const int lane = threadIdx.x & 31;
    const int gw   = (blockIdx.x * blockDim.x + threadIdx.x) >> 5;
    const int nw   = (gridDim.x * blockDim.x) >> 5;
    for (int e = gw; e < E; e += nw) {
        const int s = src[e];
        const int t = dst[e];
        const float norm = dinv[s] * ew[e] * dinv[t];
        const float4 hv = *reinterpret_cast<const float4*>(h + (size_t)s * D_FEAT + lane * 4);
        float* op = out + (size_t)t * D_FEAT + lane * 4;
        atomicAdd(op + 0, hv.x * norm);
        atomicAdd(op + 1, hv.y * norm);
        atomicAdd(op + 2, hv.z * norm);
        atomicAdd(op + 3, hv.w * norm);
    }
}

extern "C" void kernel_launch(void* const* d_in, const int* in_sizes, int n_in,
                              void* d_out, int out_size, void* d_ws, size_t ws_size,
                              hipStream_t stream) {
    const float* x  = (const float*)d_in[0];   // [N,128]
    const float* W  = (const float*)d_in[1];   // [128,128]
    const float* bb = (const float*)d_in[2];   // [128]
    const float* ew = (const float*)d_in[3];   // [E]
    const int*   ei = (const int*)d_in[4];     // [2,E]

    const int E = in_sizes[3];
    const int N = in_sizes[0] / D_FEAT;
    const int* src = ei;
    const int* dst = ei + E;
    float* out = (float*)d_out;

    // workspace layout: deg/dinv [N] | h [N*128]
    float* deg = (float*)d_ws;
    size_t degPad = ((size_t)N + 63) & ~(size_t)63;
    float* h = deg + degPad;

    const int TB = 256;
    k_init_deg <<<(N + TB - 1) / TB, TB, 0, stream>>>(deg, N);
    k_deg_accum<<<(E + TB - 1) / TB, TB, 0, stream>>>(deg, dst, ew, E);
    k_dinv     <<<(N + TB - 1) / TB, TB, 0, stream>>>(deg, N);

    // GEMM: 8 waves/block (one per 16-col tile), one block per 16 rows
    k_gemm_wmma<<<(N + 15) / 16, 256, 0, stream>>>(x, W, h, N);

    const int total = N * D_FEAT;
    k_out_init<<<(total + TB - 1) / TB, TB, 0, stream>>>(out, h, deg, bb, total);

    k_scatter<<<4096, TB, 0, stream>>>(out, h, deg, src, dst, ew, E);
}